// TO_layer_4028679323830
// MI455X (gfx1250) — compile-verified
//
#include <hip/hip_runtime.h>

typedef __attribute__((ext_vector_type(2))) float v2f;
typedef __attribute__((ext_vector_type(8))) float v8f;

constexpr int kB = 64;
constexpr int kM = 512;
constexpr int kStrips = kM / 16;          // 32 d-strips per batch
constexpr int kSplit = 4;                 // i-range split for parallelism
constexpr int kRowsPerSplit = kM / kSplit;

// ---------------------------------------------------------------------------
// Phase 1: partial[s,b,d] = sum_{i in split s, i+d<M} (L[b,i,i+d] + rho*T[b,i,i+d])
// computed as ones(16x4) x G(4x16) accumulated with V_WMMA_F32_16X16X4_F32.
// One wave = one (split s, batch b, 16-wide d-strip). The i-loop stops at the
// wave-uniform bound M - d0, skipping the fully-masked triangle (halves reads).
// ---------------------------------------------------------------------------
__global__ __launch_bounds__(128) void adj_wmma_kernel(
    const float* __restrict__ Lre, const float* __restrict__ Lim,
    const float* __restrict__ Tre, const float* __restrict__ Tim,
    const float* __restrict__ rho_p,
    float2* __restrict__ partial)         // [kSplit][kB][kM]
{
    const int lane      = threadIdx.x & 31;
    const int waveInBlk = threadIdx.x >> 5;
    const int wave      = blockIdx.x * (blockDim.x >> 5) + waveInBlk;

    const int s   = wave / (kB * kStrips);
    const int rem = wave % (kB * kStrips);
    const int b   = rem / kStrips;
    const int d0  = (rem % kStrips) * 16;

    const float rho = rho_p[0];
    const size_t base = (size_t)b * kM * kM;

    // B-matrix (4x16) lane mapping, mirroring the documented 16x4 A layout:
    //   VGPR0: lanes 0-15 -> K=0, lanes 16-31 -> K=2
    //   VGPR1: lanes 0-15 -> K=1, lanes 16-31 -> K=3
    const int n  = lane & 15;
    const int kA = (lane < 16) ? 0 : 2;

    v8f c_re = {};
    v8f c_im = {};
    const v2f ones = {1.0f, 1.0f};

    // Wave-uniform loop bounds (EXEC stays all-ones for every WMMA).
    const int iBeg = s * kRowsPerSplit;
    const int iLim = kM - d0;             // rows >= iLim are fully masked
    const int iEnd = (iBeg + kRowsPerSplit < iLim) ? (iBeg + kRowsPerSplit) : iLim;

    for (int i0 = iBeg; i0 < iEnd; i0 += 4) {
        const int i_0 = i0 + kA;
        const int i_1 = i0 + kA + 1;
        const int j_0 = i_0 + d0 + n;              // column for element (K=kA,   N=n)
        const int j_1 = i_1 + d0 + n;              // column for element (K=kA+1, N=n)

        // Branch-free edge masking: clamp address in-bounds, multiply by 0/1.
        const float m0  = (j_0 < kM) ? 1.0f : 0.0f;
        const float m1  = (j_1 < kM) ? 1.0f : 0.0f;
        const int   jc0 = (j_0 < kM) ? j_0 : (kM - 1);
        const int   jc1 = (j_1 < kM) ? j_1 : (kM - 1);
        const size_t o0 = base + (size_t)i_0 * kM + jc0;
        const size_t o1 = base + (size_t)i_1 * kM + jc1;

        v2f bre, bim;
        bre.x = m0 * fmaf(rho, Tre[o0], Lre[o0]);
        bre.y = m1 * fmaf(rho, Tre[o1], Lre[o1]);
        bim.x = m0 * fmaf(rho, Tim[o0], Lim[o0]);
        bim.y = m1 * fmaf(rho, Tim[o1], Lim[o1]);

        // (neg_a, A, neg_b, B, c_mod, C, reuse_a, reuse_b) — 8-arg pattern.
        c_re = __builtin_amdgcn_wmma_f32_16x16x4_f32(
            false, ones, false, bre, (short)0, c_re, false, false);
        c_im = __builtin_amdgcn_wmma_f32_16x16x4_f32(
            false, ones, false, bim, (short)0, c_im, false, false);
    }

    // Row M=0 of C lives in VGPR0 of lanes 0..15: the 16 column sums.
    if (lane < 16) {
        partial[(size_t)(s * kB + b) * kM + d0 + n] = make_float2(c_re[0], c_im[0]);
    }
}

// ---------------------------------------------------------------------------
// Reduce the kSplit partials, apply the e1 (tau) term and the
// 1/((M-d)*rho) normalizer: u[b,d].
// ---------------------------------------------------------------------------
__global__ __launch_bounds__(256) void finalize_u_kernel(
    const float2* __restrict__ partial, const float* __restrict__ rho_p,
    const float* __restrict__ tau_p, float2* __restrict__ u)
{
    const int idx = blockIdx.x * blockDim.x + threadIdx.x;   // b*kM + d
    if (idx >= kB * kM) return;
    const int d = idx & (kM - 1);

    float re = 0.0f, im = 0.0f;
    for (int s = 0; s < kSplit; ++s) {
        const float2 p = partial[(size_t)s * kB * kM + idx];
        re += p.x;
        im += p.y;
    }
    if (d == 0) re -= 0.5f * tau_p[0] * (float)kM;           // e1 term
    const float scale = 1.0f / ((float)(kM - d) * rho_p[0]);
    u[idx] = make_float2(re * scale, im * scale);
}

// ---------------------------------------------------------------------------
// Phase 2: T[b,i,j] = u[b, j-i] (j >= i) else conj(u[b, i-j]).
// One block per output row; each thread emits two complex elements as one
// 16-byte float4 store (coalesced b128). u[b,*] is 4 KB, cache-resident.
// ---------------------------------------------------------------------------
__global__ __launch_bounds__(256) void toeplitz_out_kernel(
    const float2* __restrict__ u, float4* __restrict__ out)
{
    const int row = blockIdx.x;            // 0 .. B*M-1
    const int b   = row / kM;
    const int i   = row % kM;
    const float2* __restrict__ ub = u + b * kM;

    const int j0 = threadIdx.x * 2;
    const int k0 = j0 - i;
    const int k1 = j0 + 1 - i;

    const float2 t0 = (k0 >= 0) ? ub[k0] : ub[-k0];
    const float2 t1 = (k1 >= 0) ? ub[k1] : ub[-k1];
    const float i0v = (k0 >= 0) ? t0.y : -t0.y;
    const float i1v = (k1 >= 0) ? t1.y : -t1.y;

    out[(size_t)row * (kM / 2) + threadIdx.x] = make_float4(t0.x, i0v, t1.x, i1v);
}

extern "C" void kernel_launch(void* const* d_in, const int* in_sizes, int n_in,
                              void* d_out, int out_size, void* d_ws, size_t ws_size,
                              hipStream_t stream) {
    const float* rho = (const float*)d_in[0];
    const float* tau = (const float*)d_in[1];
    const float* Lre = (const float*)d_in[2];
    const float* Lim = (const float*)d_in[3];
    const float* Tre = (const float*)d_in[4];
    const float* Tim = (const float*)d_in[5];
    // d_in[6] is M on device; shapes are compile-time constants here.

    float2* partial = (float2*)d_ws;                         // 1 MiB
    float2* u       = partial + (size_t)kSplit * kB * kM;    // +256 KiB

    const int wavesTotal    = kSplit * kB * kStrips;         // 8192 waves
    const int wavesPerBlock = 4;                             // 128-thread blocks
    adj_wmma_kernel<<<dim3(wavesTotal / wavesPerBlock), 32 * wavesPerBlock, 0, stream>>>(
        Lre, Lim, Tre, Tim, rho, partial);

    finalize_u_kernel<<<dim3((kB * kM + 255) / 256), 256, 0, stream>>>(
        partial, rho, tau, u);

    toeplitz_out_kernel<<<dim3(kB * kM), 256, 0, stream>>>(u, (float4*)d_out);
}